// HANHeteroGNN_5686536699974
// MI455X (gfx1250) — compile-verified
//
#include <hip/hip_runtime.h>
#include <math.h>

#define N_ADDR 150000
#define N_TX   200000
#define HID    32
#define HEADS  4
#define DIMH   8
#define E_AT   1000000
#define E_TA   1000000
#define E_AA   500000
#define SLOPE  0.2f

typedef __attribute__((ext_vector_type(2))) float v2f;
typedef __attribute__((ext_vector_type(8))) float v8f;

#define CDIV(a, b) (((a) + (b) - 1) / (b))

// ---------------------------------------------------------------------------
// float atomic max via signed/unsigned integer atomics (IEEE ordering trick)
// ---------------------------------------------------------------------------
static __device__ __forceinline__ void atomicMaxF(float* addr, float v) {
  if (v >= 0.0f)
    atomicMax((int*)addr, __float_as_int(v));
  else
    atomicMin((unsigned int*)addr, __float_as_uint(v));
}

static __device__ __forceinline__ float lrelu(float l) {
  return (l >= 0.0f) ? l : SLOPE * l;
}

// ---------------------------------------------------------------------------
// fill
// ---------------------------------------------------------------------------
__global__ void k_fill(float* __restrict__ p, float v, int n) {
  int t = blockIdx.x * blockDim.x + threadIdx.x;
  if (t < n) p[t] = v;
}

// ---------------------------------------------------------------------------
// Dense projection H = X(N x K) @ W(K x 32) + b using V_WMMA_F32_16X16X4_F32.
// One wave (32 lanes) computes one 16x16 tile of H. grid = (ceil(N/16), 2).
// A layout (16x4 f32, ISA 7.12.2): lanes 0-15 hold M=0..15; v0 = {K=0 | K=2},
// v1 = {K=1 | K=3} across lane halves. B (4x16): rows striped across lanes,
// v0 = rows {0|2}, v1 = rows {1|3}. C/D: VGPR j = rows {j | j+8}, N = lane%16.
// OOB A rows are clamped (their products land only in rows the store guard
// drops), keeping the inner loop branch-free.
// ---------------------------------------------------------------------------
__global__ void k_proj_wmma(const float* __restrict__ X, const float* __restrict__ W,
                            const float* __restrict__ bias, float* __restrict__ H,
                            int N, int K) {
  const int lane = threadIdx.x;          // 0..31, one wave per block
  const int half = lane >> 4;            // 0 or 1
  const int l16  = lane & 15;
  const int row  = blockIdx.x * 16 + l16;            // A-matrix row (logical)
  const int rl   = (row < N) ? row : (N - 1);        // clamped load row
  const int col  = blockIdx.y * 16 + l16;            // B/C/D column

  const float* __restrict__ xp = X + (size_t)rl * K + (half << 1);

  v8f c = {};
  for (int k0 = 0; k0 < K; k0 += 4) {
    const int ka = k0 + (half << 1);
    v2f a, b;
    a.x = xp[k0];
    a.y = xp[k0 + 1];
    b.x = W[ka * HID + col];
    b.y = W[(ka + 1) * HID + col];
    c = __builtin_amdgcn_wmma_f32_16x16x4_f32(false, a, false, b, (short)0, c,
                                              false, false);
  }
  const float bb = bias[col];
#pragma unroll
  for (int j = 0; j < 8; ++j) {
    int r = blockIdx.x * 16 + j + (half ? 8 : 0);
    if (r < N) H[r * HID + col] = c[j] + bb;
  }
}

// ---------------------------------------------------------------------------
// per-node, per-head attention coefficient: out[n*4+h] = <H[n,h,:], att[h,:]>
// ---------------------------------------------------------------------------
__global__ void k_node_dot(const float* __restrict__ H, const float* __restrict__ att,
                           float* __restrict__ out, int N) {
  int t = blockIdx.x * blockDim.x + threadIdx.x;
  if (t >= N * HEADS) return;
  int n = t >> 2, h = t & 3;
  const float* hp = H + n * HID + h * DIMH;
  const float* ap = att + h * DIMH;
  float v = 0.0f;
#pragma unroll
  for (int d = 0; d < DIMH; ++d) v += hp[d] * ap[d];
  out[t] = v;
}

// ---------------------------------------------------------------------------
// edge pass 1: leaky-relu logit -> segment max over destinations
// one thread per edge; alpha vectors gathered as float4 (b128)
// ---------------------------------------------------------------------------
__global__ void k_edge_max(const int* __restrict__ src, const int* __restrict__ dst,
                           const float4* __restrict__ as4, const float4* __restrict__ ad4,
                           float* __restrict__ m, int E) {
  int e = blockIdx.x * blockDim.x + threadIdx.x;
  if (e >= E) return;
  int si = src[e], di = dst[e];
  float4 a = as4[si];
  float4 d = ad4[di];
  float* mp = m + di * HEADS;
  atomicMaxF(mp + 0, lrelu(a.x + d.x));
  atomicMaxF(mp + 1, lrelu(a.y + d.y));
  atomicMaxF(mp + 2, lrelu(a.z + d.z));
  atomicMaxF(mp + 3, lrelu(a.w + d.w));
}

// ---------------------------------------------------------------------------
// edge pass 2: e = exp(logit - m[dst]) cached per edge; segment sum
// one thread per edge; float4 loads, one b128 store of exp values
// ---------------------------------------------------------------------------
__global__ void k_edge_expsum(const int* __restrict__ src, const int* __restrict__ dst,
                              const float4* __restrict__ as4, const float4* __restrict__ ad4,
                              const float4* __restrict__ m4, float* __restrict__ s,
                              float4* __restrict__ ee4, int E) {
  int e = blockIdx.x * blockDim.x + threadIdx.x;
  if (e >= E) return;
  int si = src[e], di = dst[e];
  float4 a = as4[si];
  float4 d = ad4[di];
  float4 mx = m4[di];
  float4 ex;
  ex.x = __expf(lrelu(a.x + d.x) - mx.x);
  ex.y = __expf(lrelu(a.y + d.y) - mx.y);
  ex.z = __expf(lrelu(a.z + d.z) - mx.z);
  ex.w = __expf(lrelu(a.w + d.w) - mx.w);
  ee4[e] = ex;
  float* sp = s + di * HEADS;
  atomicAdd(sp + 0, ex.x);
  atomicAdd(sp + 1, ex.y);
  atomicAdd(sp + 2, ex.z);
  atomicAdd(sp + 3, ex.w);
}

// ---------------------------------------------------------------------------
// edge pass 3: out[dst] += alpha * h_src[src]  (L2-resident f32 atomics)
// one thread per (edge, head): 32B contiguous gather + 8 f32 atomics
// ---------------------------------------------------------------------------
__global__ void k_edge_scatter(const int* __restrict__ src, const int* __restrict__ dst,
                               const float* __restrict__ H, const float* __restrict__ s,
                               const float* __restrict__ ee, float* __restrict__ out,
                               int E) {
  int t = blockIdx.x * blockDim.x + threadIdx.x;
  if (t >= E * HEADS) return;
  int e = t >> 2, h = t & 3;
  int si = src[e], di = dst[e];
  float w = ee[t] / (s[di * HEADS + h] + 1e-16f);
  const float* hp = H + si * HID + h * DIMH;
  float* op = out + di * HID + h * DIMH;
#pragma unroll
  for (int d = 0; d < DIMH; ++d) atomicAdd(&op[d], hp[d] * w);
}

__global__ void k_relu(float* __restrict__ p, int n) {
  int t = blockIdx.x * blockDim.x + threadIdx.x;
  if (t < n) p[t] = fmaxf(p[t], 0.0f);
}

// ---------------------------------------------------------------------------
// semantic attention: tsum[f] += sum_rows tanh( (x @ kW + kb)[f] )
// block = 32 features x 8 rows, grid-stride over rows, LDS reduce -> atomicAdd
// ---------------------------------------------------------------------------
__global__ void k_sem_reduce(const float* __restrict__ X, const float* __restrict__ kW,
                             const float* __restrict__ kb, float* __restrict__ tsum,
                             int N) {
  __shared__ float sh[256];
  const int f  = threadIdx.x & 31;
  const int rs = threadIdx.x >> 5;       // 0..7
  float acc = 0.0f;
  for (int r = blockIdx.x * 8 + rs; r < N; r += gridDim.x * 8) {
    const float* x = X + r * HID;
    float y = kb[f];
#pragma unroll
    for (int g = 0; g < HID; ++g) y += x[g] * kW[g * HID + f];
    acc += tanhf(y);
  }
  sh[threadIdx.x] = acc;
  __syncthreads();
  if (rs == 0) {
    float v = 0.0f;
#pragma unroll
    for (int i = 0; i < 8; ++i) v += sh[i * 32 + f];
    atomicAdd(&tsum[f], v);
  }
}

// softmax over 2 relation scores (addr side); tx side is softmax over 1 == id
__global__ void k_sem_attn2(const float* __restrict__ tsum, const float* __restrict__ q,
                            float* __restrict__ attn, float invN) {
  if (threadIdx.x == 0 && blockIdx.x == 0) {
    float s0 = 0.0f, s1 = 0.0f;
    for (int f = 0; f < HID; ++f) {
      s0 += tsum[f] * invN * q[f];
      s1 += tsum[HID + f] * invN * q[f];
    }
    float mx = fmaxf(s0, s1);
    float e0 = __expf(s0 - mx), e1 = __expf(s1 - mx);
    float inv = 1.0f / (e0 + e1);
    attn[0] = e0 * inv;
    attn[1] = e1 * inv;
  }
}

__global__ void k_sem_mix(const float* __restrict__ o1, const float* __restrict__ o2,
                          const float* __restrict__ attn, float* __restrict__ out, int n) {
  int t = blockIdx.x * blockDim.x + threadIdx.x;
  if (t < n) out[t] = attn[0] * o1[t] + attn[1] * o2[t];
}

// ---------------------------------------------------------------------------
// final linear: out[n, 0:2] = a[n, :] @ linW + linb
// ---------------------------------------------------------------------------
__global__ void k_final_lin(const float* __restrict__ A, const float* __restrict__ W,
                            const float* __restrict__ b, float* __restrict__ out, int N) {
  int t = blockIdx.x * blockDim.x + threadIdx.x;
  if (t >= N) return;
  const float* a = A + t * HID;
  float o0 = b[0], o1 = b[1];
#pragma unroll
  for (int f = 0; f < HID; ++f) {
    float x = a[f];
    o0 += x * W[f * 2];
    o1 += x * W[f * 2 + 1];
  }
  out[t * 2]     = o0;
  out[t * 2 + 1] = o1;
}

// ---------------------------------------------------------------------------
// host orchestration
// ---------------------------------------------------------------------------
extern "C" void kernel_launch(void* const* d_in, const int* in_sizes, int n_in,
                              void* d_out, int out_size, void* d_ws, size_t ws_size,
                              hipStream_t stream) {
  const float* x_addr   = (const float*)d_in[0];
  const float* x_tx     = (const float*)d_in[1];
  const int*   e_at_src = (const int*)d_in[2];
  const int*   e_at_dst = (const int*)d_in[3];
  const int*   e_ta_src = (const int*)d_in[4];
  const int*   e_ta_dst = (const int*)d_in[5];
  const int*   e_aa_src = (const int*)d_in[6];
  const int*   e_aa_dst = (const int*)d_in[7];
  const float* pW1      = (const float*)d_in[8];   // [2][64][32]
  const float* pb1      = (const float*)d_in[9];   // [2][32]
  const float* pW23     = (const float*)d_in[10];  // [2][2][32][32]
  const float* pb23     = (const float*)d_in[11];  // [2][2][32]
  const float* att_src  = (const float*)d_in[12];  // [3][3][4][8]
  const float* att_dst  = (const float*)d_in[13];
  const float* kW       = (const float*)d_in[14];  // [3][32][32]
  const float* kb       = (const float*)d_in[15];  // [3][32]
  const float* q        = (const float*)d_in[16];  // [3][32]
  const float* linW     = (const float*)d_in[17];  // [32][2]
  const float* linb     = (const float*)d_in[18];  // [2]
  float* out = (float*)d_out;

  // ---- workspace carve (256B aligned blocks) ----
  char* wp = (char*)d_ws;
  auto alloc = [&](size_t nfl) -> float* {
    float* p = (float*)wp;
    wp += ((nfl * sizeof(float) + 255) / 256) * 256;
    return p;
  };
  float* h_a   = alloc((size_t)N_ADDR * HID);
  float* h_t   = alloc((size_t)N_TX * HID);
  float* o_tx  = alloc((size_t)N_TX * HID);
  float* o_a1  = alloc((size_t)N_ADDR * HID);
  float* o_a2  = alloc((size_t)N_ADDR * HID);
  float* a_cur = alloc((size_t)N_ADDR * HID);
  float* asrc  = alloc((size_t)N_TX * HEADS);
  float* adst  = alloc((size_t)N_TX * HEADS);
  float* m_t   = alloc((size_t)N_TX * HEADS);
  float* s_t   = alloc((size_t)N_TX * HEADS);
  float* m_a   = alloc((size_t)N_ADDR * HEADS);
  float* s_a   = alloc((size_t)N_ADDR * HEADS);
  float* ee    = alloc((size_t)E_AT * HEADS);   // largest relation, reused
  float* tsum  = alloc(64);
  float* attn  = alloc(2);

  const int TB = 256;
  auto fill = [&](float* p, float v, int n) {
    k_fill<<<CDIV(n, TB), TB, 0, stream>>>(p, v, n);
  };

  // one relation-level attention pass
  auto relation = [&](const float* Hs, int Ns, const float* Hd, int Nd,
                      const int* es, const int* ed, int E,
                      const float* att_s, const float* att_d,
                      float* m, float* s, float* ob) {
    k_node_dot<<<CDIV(Ns * HEADS, TB), TB, 0, stream>>>(Hs, att_s, asrc, Ns);
    k_node_dot<<<CDIV(Nd * HEADS, TB), TB, 0, stream>>>(Hd, att_d, adst, Nd);
    fill(m, -INFINITY, Nd * HEADS);
    fill(s, 0.0f, Nd * HEADS);
    fill(ob, 0.0f, Nd * HID);
    k_edge_max<<<CDIV(E, TB), TB, 0, stream>>>(es, ed, (const float4*)asrc,
                                               (const float4*)adst, m, E);
    k_edge_expsum<<<CDIV(E, TB), TB, 0, stream>>>(es, ed, (const float4*)asrc,
                                                  (const float4*)adst, (const float4*)m,
                                                  s, (float4*)ee, E);
    k_edge_scatter<<<CDIV(E * HEADS, TB), TB, 0, stream>>>(es, ed, Hs, s, ee, ob, E);
    k_relu<<<CDIV(Nd * HID, TB), TB, 0, stream>>>(ob, Nd * HID);
  };

  // one HAN layer; new addr features -> a_cur, new tx features alias o_tx
  auto layer = [&](const float* xa, int Ka, const float* xt, int Kt,
                   const float* Wa, const float* ba, const float* Wt, const float* bt,
                   const float* aS, const float* aD,
                   const float* kWl, const float* kbl, const float* ql) {
    dim3 ga(CDIV(N_ADDR, 16), 2), gt(CDIV(N_TX, 16), 2);
    k_proj_wmma<<<ga, 32, 0, stream>>>(xa, Wa, ba, h_a, N_ADDR, Ka);
    k_proj_wmma<<<gt, 32, 0, stream>>>(xt, Wt, bt, h_t, N_TX, Kt);
    // r0: addr -> tx ; r1: tx -> addr ; r2: addr -> addr
    relation(h_a, N_ADDR, h_t, N_TX, e_at_src, e_at_dst, E_AT, aS + 0, aD + 0, m_t, s_t, o_tx);
    relation(h_t, N_TX, h_a, N_ADDR, e_ta_src, e_ta_dst, E_TA, aS + 32, aD + 32, m_a, s_a, o_a1);
    relation(h_a, N_ADDR, h_a, N_ADDR, e_aa_src, e_aa_dst, E_AA, aS + 64, aD + 64, m_a, s_a, o_a2);
    // semantic attention over the two addr relations (tx: softmax over 1 == id)
    fill(tsum, 0.0f, 64);
    k_sem_reduce<<<256, 256, 0, stream>>>(o_a1, kWl, kbl, tsum, N_ADDR);
    k_sem_reduce<<<256, 256, 0, stream>>>(o_a2, kWl, kbl, tsum + 32, N_ADDR);
    k_sem_attn2<<<1, 32, 0, stream>>>(tsum, ql, attn, 1.0f / (float)N_ADDR);
    k_sem_mix<<<CDIV(N_ADDR * HID, TB), TB, 0, stream>>>(o_a1, o_a2, attn, a_cur,
                                                         N_ADDR * HID);
  };

  // layer 1 (IN_DIM = 64)
  layer(x_addr, 64, x_tx, 64,
        pW1, pb1, pW1 + 64 * HID, pb1 + HID,
        att_src, att_dst, kW, kb, q);
  // layers 2 and 3 (HID -> HID); wrapper relu is identity (outputs nonneg)
  for (int l = 0; l < 2; ++l) {
    layer(a_cur, HID, o_tx, HID,
          pW23 + (l * 2 + 0) * HID * HID, pb23 + (l * 2 + 0) * HID,
          pW23 + (l * 2 + 1) * HID * HID, pb23 + (l * 2 + 1) * HID,
          att_src + (l + 1) * 96, att_dst + (l + 1) * 96,
          kW + (l + 1) * HID * HID, kb + (l + 1) * HID, q + (l + 1) * HID);
  }

  k_final_lin<<<CDIV(N_ADDR, TB), TB, 0, stream>>>(a_cur, linW, linb, out, N_ADDR);
}